// SS2D_7009386627566
// MI455X (gfx1250) — compile-verified
//
#include <hip/hip_runtime.h>
#include <math.h>

// ---------------------------------------------------------------------------
// SS2D (VMamba selective-scan 2D) forward for gfx1250 (MI455X).
// - fp32 WMMA (v_wmma_f32_16x16x4_f32) for all projection GEMMs (full precision,
//   matters because errors feed exp() through a 4096-step recurrence).
// - Selective scan: fused dt_proj+softplus, register-resident 16-wide state,
//   x_dbl descriptor stream staged into LDS with double-buffered
//   global_load_async_to_lds_b128 (ASYNCcnt) so the wave-uniform loads come
//   off the recurrence critical path.
// - Working set < 100 MB -> L2 (192 MB) resident; bottleneck is the L=4096
//   sequential recurrence (only 1536 independent rows = 48 wave32 waves), so
//   the design minimizes per-step latency rather than chasing bandwidth.
// ---------------------------------------------------------------------------

typedef __attribute__((ext_vector_type(2))) float v2f;
typedef __attribute__((ext_vector_type(8))) float v8f;
typedef __attribute__((ext_vector_type(4))) int   v4i;

// pointer-to-AS-qualified-int4 aliases for the async-copy builtin
using gv4i_p = v4i __attribute__((address_space(1))) *;   // global int4*
using lv4i_p = v4i __attribute__((address_space(3))) *;   // LDS int4*

#define BB   2
#define HH   64
#define WW   64
#define LL   4096          // H*W
#define DM   96            // D_MODEL
#define DI   192           // D_INNER
#define NST  16            // D_STATE
#define DTR  6             // DT_RANK
#define KD   4             // scan directions
#define XZC  384           // 2*D_INNER
#define JP   48            // padded per-position stride of x_dbl (dts@0..5, B@8..23, C@24..39)
#define MROWS (BB*LL)      // 8192 flattened (b,h,w) rows
#define CH   32            // scan steps per LDS-staged chunk
#define NCH  (LL/CH)       // 128 chunks

#if __has_builtin(__builtin_amdgcn_global_load_async_to_lds_b128) && \
    __has_builtin(__builtin_amdgcn_s_wait_asynccnt)
#define HAVE_ASYNC 1
#else
#define HAVE_ASYNC 0
#endif

__device__ __forceinline__ float sigmoidf_(float x) { return 1.0f / (1.0f + __expf(-x)); }
__device__ __forceinline__ float softplusf_(float x) { return (x > 20.0f) ? x : log1pf(__expf(x)); }

// spatial position for scan-step l of direction k
__device__ __forceinline__ int pos_of(int k, int l) {
    switch (k & 3) {
        case 0: return l;                                   // hw order
        case 1: return ((l & 63) << 6) | (l >> 6);          // wh order: l = w*H + h
        case 2: return (LL - 1) - l;                        // hw reversed
        default: { int r = (LL - 1) - l; return ((r & 63) << 6) | (r >> 6); }
    }
}

// Stage one chunk (CH*JP = 1536 floats = 6 KB) of the x_dbl stream into LDS.
// 192 threads x 2 x b128 covers it exactly. Async path tracked by ASYNCcnt.
__device__ __forceinline__ void stage_chunk(float* dst, const float* src, int tid) {
#if HAVE_ASYNC
    __builtin_amdgcn_global_load_async_to_lds_b128(
        (gv4i_p)(src + tid * 4), (lv4i_p)(dst + tid * 4), 0, 0);
    __builtin_amdgcn_global_load_async_to_lds_b128(
        (gv4i_p)(src + (tid + DI) * 4), (lv4i_p)(dst + (tid + DI) * 4), 0, 0);
#else
    const float4 a = *(const float4*)(src + tid * 4);
    const float4 b = *(const float4*)(src + (tid + DI) * 4);
    *(float4*)(dst + tid * 4) = a;
    *(float4*)(dst + (tid + DI) * 4) = b;
#endif
}

// ---------------------------------------------------------------------------
// Kernel 1: in_proj  xz[m, o] = sum_c x[m, c] * ipw[o, c]   (8192 x 96 @ 96 x 384)
// One wave per 16x16 C tile; fp32 WMMA, K-loop step 4.
// A frag layout (16x4 f32): lane<16 holds K=0,1 of row M=lane; lane>=16 K=2,3 of M=lane-16.
// B frag mirrors A's K split (K0/K2 in v0, K1/K3 in v1, N = lane&15).
// C/D layout: VGPR j: lanes 0-15 -> M=j, lanes 16-31 -> M=j+8; N = lane&15.
// ---------------------------------------------------------------------------
__global__ __launch_bounds__(32) void k_inproj(const float* __restrict__ x,
                                               const float* __restrict__ ipw,
                                               float* __restrict__ xz) {
    const int lane = threadIdx.x, lm = lane & 15, koff = (lane >> 4) * 2;
    const int n0 = blockIdx.x * 16, m0 = blockIdx.y * 16;
    const float* arow = x   + (size_t)(m0 + lm) * DM;
    const float* brow = ipw + (size_t)(n0 + lm) * DM;   // B[kc, o] = ipw[o, kc]
    v8f acc = {0.f, 0.f, 0.f, 0.f, 0.f, 0.f, 0.f, 0.f};
    for (int kk = 0; kk < DM; kk += 4) {
        v2f a  = { arow[kk + koff], arow[kk + koff + 1] };
        v2f bf = { brow[kk + koff], brow[kk + koff + 1] };
        acc = __builtin_amdgcn_wmma_f32_16x16x4_f32(false, a, false, bf, (short)0, acc, false, false);
    }
    const int rbase = m0 + ((lane >> 4) << 3);
    #pragma unroll
    for (int j = 0; j < 8; j++)
        xz[(size_t)(rbase + j) * XZC + n0 + lm] = acc[j];
}

// ---------------------------------------------------------------------------
// Kernel 2: 3x3 depthwise conv (SAME) + bias + SiLU.  xin = xz[:, 0:192] (NHWC),
// output xc laid out (B, C, L) channel-major for the scan.
// ---------------------------------------------------------------------------
__global__ void k_conv(const float* __restrict__ xz,
                       const float* __restrict__ cw,
                       const float* __restrict__ cb,
                       float* __restrict__ xc) {
    int idx = blockIdx.x * blockDim.x + threadIdx.x;
    if (idx >= BB * DI * LL) return;
    const int l = idx & (LL - 1);
    const int c = (idx >> 12) % DI;
    const int b = idx / (DI * LL);
    const int h = l >> 6, w = l & 63;
    float acc = cb[c];
    #pragma unroll
    for (int ky = 0; ky < 3; ky++) {
        const int h2 = h + ky - 1;
        if ((unsigned)h2 >= (unsigned)HH) continue;
        #pragma unroll
        for (int kx = 0; kx < 3; kx++) {
            const int w2 = w + kx - 1;
            if ((unsigned)w2 >= (unsigned)WW) continue;
            acc = fmaf(cw[c * 9 + ky * 3 + kx],
                       xz[((size_t)b * LL + h2 * WW + w2) * XZC + c], acc);
        }
    }
    acc = acc * sigmoidf_(acc);   // SiLU
    xc[((size_t)b * DI + c) * LL + l] = acc;
}

// ---------------------------------------------------------------------------
// Kernel 3: x_proj per direction.  x_dbl[b,k,j,l] = sum_c xs_k[b,c,l] * xpw[k,c,j]
// GEMM M=8192 (b,l), N=38 (padded to 48), K=192. A gathered from xc via pos_of.
// Output stored padded: j<6 -> slot j (dts), 6..21 -> slot j+2 (B), 22..37 -> j+2 (C)
// so the scan can use 16B-aligned vector loads.
// ---------------------------------------------------------------------------
__global__ __launch_bounds__(32) void k_xproj(const float* __restrict__ xc,
                                              const float* __restrict__ xpw,
                                              float* __restrict__ xdbl) {
    const int lane = threadIdx.x, lm = lane & 15, koff = (lane >> 4) * 2;
    const int n0 = blockIdx.x * 16, m0 = blockIdx.y * 16, k = blockIdx.z;
    const int m = m0 + lm, b = m >> 12, l = m & (LL - 1);
    const int pos = pos_of(k, l);
    const float* abase = xc + (size_t)b * DI * LL + pos;   // stride LL per channel
    const float* wbase = xpw + (size_t)k * DI * 38;
    const int jcol = n0 + lm;
    v8f acc = {0.f, 0.f, 0.f, 0.f, 0.f, 0.f, 0.f, 0.f};
    for (int kk = 0; kk < DI; kk += 4) {
        const int c = kk + koff;
        v2f a = { abase[(size_t)c * LL], abase[(size_t)(c + 1) * LL] };
        const float b0 = (jcol < 38) ? wbase[c * 38 + jcol] : 0.0f;
        const float b1 = (jcol < 38) ? wbase[(c + 1) * 38 + jcol] : 0.0f;
        v2f bf = { b0, b1 };
        acc = __builtin_amdgcn_wmma_f32_16x16x4_f32(false, a, false, bf, (short)0, acc, false, false);
    }
    if (jcol < 38) {
        const int jj = (jcol < 6) ? jcol : jcol + 2;       // padded slot
        const int rbase = m0 + ((lane >> 4) << 3);
        #pragma unroll
        for (int jr = 0; jr < 8; jr++) {
            const int mm = rbase + jr;
            const int bb = mm >> 12, ll2 = mm & (LL - 1);
            xdbl[((size_t)(bb * KD + k) * LL + ll2) * JP + jj] = acc[jr];
        }
    }
}

// ---------------------------------------------------------------------------
// Kernel 4: selective scan, fused dt_proj + softplus + D-skip.
// One 192-thread block (6 waves) per (b, k); thread owns channel d with all 16
// states in VGPRs. The shared x_dbl line stream is staged into LDS in 32-step
// chunks, double-buffered with global_load_async_to_lds_b128 + s_wait_asynccnt,
// so per-step descriptor reads are LDS broadcasts instead of redundant
// wave-uniform global loads on the recurrence critical path.
// ---------------------------------------------------------------------------
__global__ __launch_bounds__(DI) void k_scan(const float* __restrict__ xc,
                                             const float* __restrict__ xdbl,
                                             const float* __restrict__ dtw_g,
                                             const float* __restrict__ dtb_g,
                                             const float* __restrict__ alogs,
                                             const float* __restrict__ dskip,
                                             float* __restrict__ ys) {
    __shared__ __align__(16) float buf[2][CH * JP];        // 2 x 6 KB double buffer
    const int d = threadIdx.x;                             // channel 0..191
    const int k = blockIdx.x, b = blockIdx.y;
    const int kd = k * DI + d;

    float An[NST];
    #pragma unroll
    for (int n = 0; n < NST; n++) An[n] = -__expf(alogs[(size_t)kd * NST + n]);
    float dtw[DTR];
    #pragma unroll
    for (int r = 0; r < DTR; r++) dtw[r] = dtw_g[(size_t)kd * DTR + r];
    const float dtb = dtb_g[kd];
    const float Dk  = dskip[kd];

    float st[NST];
    #pragma unroll
    for (int n = 0; n < NST; n++) st[n] = 0.0f;

    const float* xcrow = xc + ((size_t)b * DI + d) * LL;
    const float* dbase = xdbl + (size_t)(b * KD + k) * LL * JP;
    float* yrow = ys + ((size_t)(b * KD + k) * DI + d) * LL;

    stage_chunk(&buf[0][0], dbase, d);                     // prologue: chunk 0 in flight

    for (int ci = 0; ci < NCH; ci++) {
        if (ci + 1 < NCH)                                  // keep next chunk in flight
            stage_chunk(&buf[(ci + 1) & 1][0], dbase + (size_t)(ci + 1) * CH * JP, d);
#if HAVE_ASYNC
        if (ci + 1 < NCH) __builtin_amdgcn_s_wait_asynccnt(2);  // chunk ci retired, ci+1 outstanding
        else              __builtin_amdgcn_s_wait_asynccnt(0);
#endif
        __syncthreads();                                   // publish chunk ci to all 6 waves

        const float* lp = &buf[ci & 1][0];
        const int l0 = ci * CH;
        for (int t = 0; t < CH; t++) {
            const int l = l0 + t;
            const float* p = lp + t * JP;                  // LDS broadcast reads
            const float4 dv0 = *(const float4*)(p);        // dts[0..3]
            const float2 dv1 = *(const float2*)(p + 4);    // dts[4..5]
            const int pos = pos_of(k, l);
            const float u = xcrow[pos];
            float dtv = dtb;
            dtv = fmaf(dtw[0], dv0.x, dtv); dtv = fmaf(dtw[1], dv0.y, dtv);
            dtv = fmaf(dtw[2], dv0.z, dtv); dtv = fmaf(dtw[3], dv0.w, dtv);
            dtv = fmaf(dtw[4], dv1.x, dtv); dtv = fmaf(dtw[5], dv1.y, dtv);
            const float delta = softplusf_(dtv);
            const float du = delta * u;
            float acc = 0.0f;
            #pragma unroll
            for (int g = 0; g < 4; g++) {
                const float4 Bv = *(const float4*)(p + 8  + 4 * g);
                const float4 Cv = *(const float4*)(p + 24 + 4 * g);
                const int n0 = g * 4;
                st[n0+0] = fmaf(__expf(delta * An[n0+0]), st[n0+0], du * Bv.x); acc = fmaf(st[n0+0], Cv.x, acc);
                st[n0+1] = fmaf(__expf(delta * An[n0+1]), st[n0+1], du * Bv.y); acc = fmaf(st[n0+1], Cv.y, acc);
                st[n0+2] = fmaf(__expf(delta * An[n0+2]), st[n0+2], du * Bv.z); acc = fmaf(st[n0+2], Cv.z, acc);
                st[n0+3] = fmaf(__expf(delta * An[n0+3]), st[n0+3], du * Bv.w); acc = fmaf(st[n0+3], Cv.w, acc);
            }
            yrow[l] = acc + u * Dk;                        // D skip-connection
        }
        __syncthreads();                                   // buffer reuse guard
    }
}

// ---------------------------------------------------------------------------
// Kernel 5: combine 4 directions (with index remap), LayerNorm over C=192,
// SiLU(z) gating.  One 192-thread block per (b, pos); LDS tree reductions.
// ---------------------------------------------------------------------------
__global__ __launch_bounds__(192) void k_combine(const float* __restrict__ ys,
                                                 const float* __restrict__ xz,
                                                 const float* __restrict__ lnw,
                                                 const float* __restrict__ lnb,
                                                 float* __restrict__ yg) {
    __shared__ float sh[DI];
    __shared__ float stats[2];
    const int c = threadIdx.x;
    const int bp = blockIdx.x;
    const int b = bp >> 12, pos = bp & (LL - 1);
    const int h = pos >> 6, w = pos & 63;
    const int l1 = (w << 6) | h;                           // wh-order step index for this pos
    const size_t base = (size_t)b * KD * DI * LL;
    float ysum = ys[base + ((size_t)0 * DI + c) * LL + pos]
               + ys[base + ((size_t)1 * DI + c) * LL + l1]
               + ys[base + ((size_t)2 * DI + c) * LL + (LL - 1 - pos)]
               + ys[base + ((size_t)3 * DI + c) * LL + (LL - 1 - l1)];

    sh[c] = ysum; __syncthreads();
    if (c < 64) sh[c] = sh[c] + sh[c + 64] + sh[c + 128];
    __syncthreads();
    for (int s = 32; s > 0; s >>= 1) { if (c < s) sh[c] += sh[c + s]; __syncthreads(); }
    if (c == 0) stats[0] = sh[0] * (1.0f / DI);
    __syncthreads();
    const float mu = stats[0];
    const float dv = ysum - mu;
    sh[c] = dv * dv; __syncthreads();
    if (c < 64) sh[c] = sh[c] + sh[c + 64] + sh[c + 128];
    __syncthreads();
    for (int s = 32; s > 0; s >>= 1) { if (c < s) sh[c] += sh[c + s]; __syncthreads(); }
    if (c == 0) stats[1] = sh[0] * (1.0f / DI);
    __syncthreads();
    const float var = stats[1];
    const float yln = dv * rsqrtf(var + 1e-5f) * lnw[c] + lnb[c];
    const float z = xz[(size_t)bp * XZC + DI + c];
    yg[(size_t)bp * DI + c] = yln * z * sigmoidf_(z);
}

// ---------------------------------------------------------------------------
// Kernel 6: out_proj  out[m, o] = sum_c yg[m, c] * opw[o, c]  (8192 x 192 @ 192 x 96)
// ---------------------------------------------------------------------------
__global__ __launch_bounds__(32) void k_outproj(const float* __restrict__ yg,
                                                const float* __restrict__ opw,
                                                float* __restrict__ out) {
    const int lane = threadIdx.x, lm = lane & 15, koff = (lane >> 4) * 2;
    const int n0 = blockIdx.x * 16, m0 = blockIdx.y * 16;
    const float* arow = yg  + (size_t)(m0 + lm) * DI;
    const float* brow = opw + (size_t)(n0 + lm) * DI;
    v8f acc = {0.f, 0.f, 0.f, 0.f, 0.f, 0.f, 0.f, 0.f};
    for (int kk = 0; kk < DI; kk += 4) {
        v2f a  = { arow[kk + koff], arow[kk + koff + 1] };
        v2f bf = { brow[kk + koff], brow[kk + koff + 1] };
        acc = __builtin_amdgcn_wmma_f32_16x16x4_f32(false, a, false, bf, (short)0, acc, false, false);
    }
    const int rbase = m0 + ((lane >> 4) << 3);
    #pragma unroll
    for (int j = 0; j < 8; j++)
        out[(size_t)(rbase + j) * DM + n0 + lm] = acc[j];
}

// ---------------------------------------------------------------------------
extern "C" void kernel_launch(void* const* d_in, const int* in_sizes, int n_in,
                              void* d_out, int out_size, void* d_ws, size_t ws_size,
                              hipStream_t stream) {
    (void)in_sizes; (void)n_in; (void)out_size; (void)ws_size;
    const float* x     = (const float*)d_in[0];
    const float* ipw   = (const float*)d_in[1];
    const float* cw    = (const float*)d_in[2];
    const float* cb    = (const float*)d_in[3];
    const float* xpw   = (const float*)d_in[4];
    const float* dtw   = (const float*)d_in[5];
    const float* dtb   = (const float*)d_in[6];
    const float* alogs = (const float*)d_in[7];
    const float* ds    = (const float*)d_in[8];
    const float* lnw   = (const float*)d_in[9];
    const float* lnb   = (const float*)d_in[10];
    const float* opw   = (const float*)d_in[11];
    float* out = (float*)d_out;

    // workspace carve-up (floats); all offsets 16B-aligned; every buffer fully
    // written before read (poison-safe)
    float* ws   = (float*)d_ws;
    float* xz   = ws;                                  // 8192*384    = 12.6 MB
    float* xc   = xz   + (size_t)MROWS * XZC;          // 2*192*4096  =  6.3 MB
    float* xdbl = xc   + (size_t)BB * DI * LL;         // 2*4*4096*48 =  6.3 MB
    float* ys   = xdbl + (size_t)BB * KD * LL * JP;    // 2*4*192*4096= 25.2 MB
    float* yg   = ys   + (size_t)BB * KD * DI * LL;    // 8192*192    =  6.3 MB

    k_inproj <<<dim3(XZC / 16, MROWS / 16),      32, 0, stream>>>(x, ipw, xz);
    k_conv   <<<(BB * DI * LL + 255) / 256,     256, 0, stream>>>(xz, cw, cb, xc);
    k_xproj  <<<dim3(3, MROWS / 16, KD),         32, 0, stream>>>(xc, xpw, xdbl);
    k_scan   <<<dim3(KD, BB),                    DI, 0, stream>>>(xc, xdbl, dtw, dtb, alogs, ds, ys);
    k_combine<<<MROWS,                           DI, 0, stream>>>(ys, xz, lnw, lnb, yg);
    k_outproj<<<dim3(DM / 16, MROWS / 16),       32, 0, stream>>>(yg, opw, out);
}